// BaseQuantizer_19232863551820
// MI455X (gfx1250) — compile-verified
//
#include <hip/hip_runtime.h>
#include <math.h>

// Problem constants (from reference)
#define N_    16384
#define D_    1024
#define G_    32
#define C_    256
#define DSUB_ 32

#define TILE_N    16
#define CB_STRIDE 36    // 32 + 4 pad: 36c mod 64 distinct for c=0..15, rows 16B-aligned
#define XG_STRIDE 36
#define SC_STRIDE 260   // 256 + 4 pad; 260*4B = 1040B rows (16B multiple)

typedef float v2f __attribute__((ext_vector_type(2)));
typedef float v8f __attribute__((ext_vector_type(8)));

// ---- CDNA5 async global<->LDS path ----
#if __has_builtin(__builtin_amdgcn_global_load_async_to_lds_b128) && \
    __has_builtin(__builtin_amdgcn_global_store_async_from_lds_b128) && \
    __has_builtin(__builtin_amdgcn_s_wait_asynccnt)
#define USE_ASYNC_LDS 1
#else
#define USE_ASYNC_LDS 0
#endif

#if USE_ASYNC_LDS
// Builtin prototype (from clang diagnostic): pointee is vector_size(16) int,
// global side in AS(1) ("__device__"), LDS side in AS(3).
typedef int b128 __attribute__((vector_size(16)));
typedef __attribute__((address_space(1))) b128* gas_b128_p;
typedef __attribute__((address_space(3))) b128* las_b128_p;

__device__ __forceinline__ void async_g2l_b128(float* lds, const float* g) {
    __builtin_amdgcn_global_load_async_to_lds_b128(
        (gas_b128_p)(g), (las_b128_p)(lds), 0, 0);
}
__device__ __forceinline__ void async_l2g_b128(float* g, const float* lds) {
    __builtin_amdgcn_global_store_async_from_lds_b128(
        (gas_b128_p)(g), (las_b128_p)(lds), 0, 0);
}
#endif

__launch_bounds__(256)
__global__ void gumbel_vq_kernel(const float* __restrict__ x,
                                 const float* __restrict__ codebook,
                                 const float* __restrict__ log_temp,
                                 const float* __restrict__ gumbel,
                                 float* __restrict__ x_recon,
                                 float* __restrict__ probs)
{
    __shared__ alignas(16) float cb_s[C_ * CB_STRIDE];     // codebook[g]
    __shared__ alignas(16) float xg_s[TILE_N * XG_STRIDE]; // x tile
    __shared__ alignas(16) float sc_s[TILE_N * SC_STRIDE]; // gumbel -> scores -> probs
    __shared__ float red_s[TILE_N * 16];                   // softmax partial reduce
    __shared__ float rowstat_s[TILE_N];                    // row max, then 1/rowsum
    __shared__ float nx_s[TILE_N];                         // ||x_ng||^2
    __shared__ float nc_s[C_];                             // ||c_gc||^2
    __shared__ float rp_s[4 * TILE_N * DSUB_];             // GEMM2 partials

    const int tid  = threadIdx.x;
    const int lane = tid & 31;
    const int wave = tid >> 5;               // 0..7
    const int g    = blockIdx.x & (G_ - 1);  // group
    const int n0   = (blockIdx.x >> 5) * TILE_N;

    const float*  cbg  = codebook + (size_t)g * C_ * DSUB_;
    const float*  gsrc = gumbel + (size_t)n0 * (G_ * C_) + (size_t)g * C_;

    // ---- stage x tile (16x32), codebook[g] (256x32), gumbel tile (16x256) ----
#if USE_ASYNC_LDS
    {
        // gumbel tile -> sc_s : 1024 x 16B chunks, 4 per thread, coalesced
        #pragma unroll
        for (int it = 0; it < 4; ++it) {
            int j = tid + it * 256;          // chunk id
            int r = j >> 6, q = j & 63;      // row, float4-in-row
            async_g2l_b128(&sc_s[r * SC_STRIDE + q * 4],
                           gsrc + (size_t)r * (G_ * C_) + q * 4);
        }
        // codebook[g] -> cb_s : 2048 chunks, 8 per thread
        #pragma unroll
        for (int it = 0; it < 8; ++it) {
            int j = tid + it * 256;
            int c = j >> 3, q = j & 7;
            async_g2l_b128(&cb_s[c * CB_STRIDE + q * 4], cbg + c * DSUB_ + q * 4);
        }
        // x tile -> xg_s : 128 chunks
        if (tid < 128) {
            int r = tid >> 3, q = tid & 7;
            async_g2l_b128(&xg_s[r * XG_STRIDE + q * 4],
                           x + (size_t)(n0 + r) * D_ + g * DSUB_ + q * 4);
        }
    }
#else
    {
        #pragma unroll
        for (int it = 0; it < 4; ++it) {
            int j = tid + it * 256;
            int r = j >> 6, q = j & 63;
            *(float4*)&sc_s[r * SC_STRIDE + q * 4] =
                *(const float4*)(gsrc + (size_t)r * (G_ * C_) + q * 4);
        }
        #pragma unroll
        for (int it = 0; it < 32; ++it) {
            int i = tid + it * 256;
            int c = i >> 5, d = i & 31;
            cb_s[c * CB_STRIDE + d] = cbg[i];
        }
        #pragma unroll
        for (int it = 0; it < 2; ++it) {
            int i = tid + it * 256;
            int r = i >> 5, c = i & 31;
            xg_s[r * XG_STRIDE + c] = x[(size_t)(n0 + r) * D_ + g * DSUB_ + c];
        }
    }
#endif
    const float tscale = __expf(-log_temp[g]);   // uniform per block
#if USE_ASYNC_LDS
    __builtin_amdgcn_s_wait_asynccnt(0);
#endif
    __syncthreads();

    // ---- squared norms ----
    {
        float s = 0.f;
        #pragma unroll
        for (int d = 0; d < DSUB_; ++d) { float v = cb_s[tid * CB_STRIDE + d]; s += v * v; }
        nc_s[tid] = s;
    }
    if (tid < TILE_N) {
        float s = 0.f;
        #pragma unroll
        for (int d = 0; d < DSUB_; ++d) { float v = xg_s[tid * XG_STRIDE + d]; s += v * v; }
        nx_s[tid] = s;
    }
    __syncthreads();

    // ---- GEMM1: scores = xg(16x32) @ cb^T(32x256), each wave -> two 16x16 tiles ----
    const int m  = lane & 15;     // A row / B col index for this lane
    const int kh = lane >> 4;     // K half-select (f32 WMMA fragment layout)
    #pragma unroll
    for (int t = 0; t < 2; ++t) {
        const int ct = wave * 2 + t;            // column tile 0..15
        v8f acc = {};
        #pragma unroll
        for (int kk = 0; kk < 8; ++kk) {
            const int k0 = kk * 4 + kh * 2;
            v2f a, b;
            a.x = xg_s[m * XG_STRIDE + k0];
            a.y = xg_s[m * XG_STRIDE + k0 + 1];
            const int cc = ct * 16 + m;         // code index (B column)
            b.x = cb_s[cc * CB_STRIDE + k0];
            b.y = cb_s[cc * CB_STRIDE + k0 + 1];
            acc = __builtin_amdgcn_wmma_f32_16x16x4_f32(
                false, a, false, b, (short)0, acc, false, false);
        }
        // fused score transform; sc_s currently holds gumbel at each slot
        #pragma unroll
        for (int j = 0; j < 8; ++j) {
            const int row = j + 8 * kh;         // C layout: reg j -> row j+8*(lane>=16)
            const int col = ct * 16 + m;
            float* slot = &sc_s[row * SC_STRIDE + col];
            *slot = (2.f * acc[j] - nx_s[row] - nc_s[col]) * tscale + *slot;
        }
    }
    __syncthreads();

    // ---- stable softmax over 256 codes per row (two-level reduce) ----
    {
        const int row = tid >> 4, seg = tid & 15;  // 16 threads/row, 16 elems each
        float mx = -3.402823466e38f;
        #pragma unroll
        for (int j = 0; j < 16; ++j)
            mx = fmaxf(mx, sc_s[row * SC_STRIDE + seg * 16 + j]);
        red_s[row * 16 + seg] = mx;
    }
    __syncthreads();
    if (tid < TILE_N) {
        float mx = red_s[tid * 16];
        #pragma unroll
        for (int j = 1; j < 16; ++j) mx = fmaxf(mx, red_s[tid * 16 + j]);
        rowstat_s[tid] = mx;
    }
    __syncthreads();
    {
        const int row = tid >> 4, seg = tid & 15;
        const float mx = rowstat_s[row];
        float s = 0.f;
        #pragma unroll
        for (int j = 0; j < 16; ++j) {
            float e = __expf(sc_s[row * SC_STRIDE + seg * 16 + j] - mx);
            sc_s[row * SC_STRIDE + seg * 16 + j] = e;
            s += e;
        }
        red_s[row * 16 + seg] = s;
    }
    __syncthreads();
    if (tid < TILE_N) {
        float s = 0.f;
        #pragma unroll
        for (int j = 0; j < 16; ++j) s += red_s[tid * 16 + j];
        rowstat_s[tid] = 1.f / s;
    }
    __syncthreads();
    // ---- normalize in LDS; write probs ----
    {
        const int row = tid >> 4, seg = tid & 15;
        const float inv = rowstat_s[row];
#if USE_ASYNC_LDS
        #pragma unroll
        for (int j = 0; j < 16; ++j)
            sc_s[row * SC_STRIDE + seg * 16 + j] *= inv;
#else
        float4* dst = (float4*)(probs + (size_t)(n0 + row) * (G_ * C_)
                                      + (size_t)g * C_ + seg * 16);
        #pragma unroll
        for (int q = 0; q < 4; ++q) {
            float4 v;
            v.x = (sc_s[row * SC_STRIDE + seg * 16 + q * 4 + 0] *= inv);
            v.y = (sc_s[row * SC_STRIDE + seg * 16 + q * 4 + 1] *= inv);
            v.z = (sc_s[row * SC_STRIDE + seg * 16 + q * 4 + 2] *= inv);
            v.w = (sc_s[row * SC_STRIDE + seg * 16 + q * 4 + 3] *= inv);
            dst[q] = v;
        }
#endif
    }
    __syncthreads();
#if USE_ASYNC_LDS
    // probs writeback streams from LDS while GEMM2 runs (completes at s_endpgm)
    {
        float* pdst = probs + (size_t)n0 * (G_ * C_) + (size_t)g * C_;
        #pragma unroll
        for (int it = 0; it < 4; ++it) {
            int j = tid + it * 256;
            int r = j >> 6, q = j & 63;
            async_l2g_b128(pdst + (size_t)r * (G_ * C_) + q * 4,
                           &sc_s[r * SC_STRIDE + q * 4]);
        }
    }
#endif

    // ---- GEMM2: recon = probs(16x256) @ cb(256x32); 2 N-tiles x 4 K-chunks ----
    {
        const int ntile = wave & 1;          // d columns 0-15 or 16-31
        const int kc    = wave >> 1;         // K chunk of 64 (0..3)
        v8f acc = {};
        #pragma unroll
        for (int kk = 0; kk < 16; ++kk) {
            const int k0 = kc * 64 + kk * 4 + kh * 2;
            v2f a, b;
            a.x = sc_s[m * SC_STRIDE + k0];
            a.y = sc_s[m * SC_STRIDE + k0 + 1];
            const int d = ntile * 16 + m;
            b.x = cb_s[k0 * CB_STRIDE + d];
            b.y = cb_s[(k0 + 1) * CB_STRIDE + d];
            acc = __builtin_amdgcn_wmma_f32_16x16x4_f32(
                false, a, false, b, (short)0, acc, false, false);
        }
        #pragma unroll
        for (int j = 0; j < 8; ++j) {
            const int row = j + 8 * kh;
            rp_s[(kc * TILE_N + row) * DSUB_ + ntile * 16 + m] = acc[j];
        }
    }
    __syncthreads();
    // reduce the 4 K-chunk partials and store x_recon (coalesced)
    #pragma unroll
    for (int it = 0; it < 2; ++it) {
        int i = tid + it * 256;              // 0..511
        int row = i >> 5, d = i & 31;
        float v = 0.f;
        #pragma unroll
        for (int kc = 0; kc < 4; ++kc) v += rp_s[(kc * TILE_N + row) * DSUB_ + d];
        x_recon[(size_t)(n0 + row) * D_ + g * DSUB_ + d] = v;
    }
}

extern "C" void kernel_launch(void* const* d_in, const int* in_sizes, int n_in,
                              void* d_out, int out_size, void* d_ws, size_t ws_size,
                              hipStream_t stream) {
    const float* x        = (const float*)d_in[0];
    const float* codebook = (const float*)d_in[1];
    const float* log_temp = (const float*)d_in[2];
    const float* gumbel   = (const float*)d_in[3];

    float* x_recon = (float*)d_out;                       // N*D floats
    float* probs   = (float*)d_out + (size_t)N_ * D_;     // N*G*C floats

    const int blocks = (N_ / TILE_N) * G_;                // 1024 * 32 = 32768
    gumbel_vq_kernel<<<blocks, 256, 0, stream>>>(x, codebook, log_temp, gumbel,
                                                 x_recon, probs);
}